// seg_decoder_19318762897750
// MI455X (gfx1250) — compile-verified
//
#include <hip/hip_runtime.h>
#include <math.h>

#define CDIV(a,b) (((a)+(b)-1)/(b))

typedef __attribute__((ext_vector_type(2))) float v2f;
typedef __attribute__((ext_vector_type(8))) float v8f;

static __device__ __forceinline__ v8f v8f_splat(float x) {
  v8f r;
#pragma unroll
  for (int i = 0; i < 8; ++i) r[i] = x;
  return r;
}

// D = A(16x4 f32) * B(4x16 f32) + C(16x16 f32)  -- CDNA5 V_WMMA_F32_16X16X4_F32
static __device__ __forceinline__ v8f wmma16x16x4(v2f a, v2f b, v8f c) {
  return __builtin_amdgcn_wmma_f32_16x16x4_f32(false, a, false, b, (short)0, c, false, false);
}

// ---------------------------------------------------------------------------
// KNN: brute force, per-thread sorted insertion (sets match jax.lax.top_k(-d))
// ---------------------------------------------------------------------------
template<int K>
__global__ void knn_kernel(const float* __restrict__ db, int ndb,
                           const float* __restrict__ q, int nq,
                           int* __restrict__ idx_out, float* __restrict__ d0_out) {
  int i = blockIdx.x * blockDim.x + threadIdx.x;
  if (i >= nq) return;
  float qx = q[3 * i], qy = q[3 * i + 1], qz = q[3 * i + 2];
  float bd[K]; int bi[K];
#pragma unroll
  for (int t = 0; t < K; ++t) { bd[t] = 3.4e38f; bi[t] = 0; }
  for (int j = 0; j < ndb; ++j) {
    float dx = db[3 * j] - qx, dy = db[3 * j + 1] - qy, dz = db[3 * j + 2] - qz;
    float d = dx * dx + dy * dy + dz * dz;
    if (d < bd[K - 1]) {
      int p = K - 1;
      while (p > 0 && bd[p - 1] > d) { bd[p] = bd[p - 1]; bi[p] = bi[p - 1]; --p; }
      bd[p] = d; bi[p] = j;
    }
  }
  for (int t = 0; t < K; ++t) idx_out[i * K + t] = bi[t];
  if (d0_out) d0_out[i] = bd[0];
}

// ---------------------------------------------------------------------------
// Deterministic FPS (start at 0, collect pre-update carry like lax.scan)
// ---------------------------------------------------------------------------
__global__ void fps_kernel(const float* __restrict__ xyz, int n, int npoint,
                           int* __restrict__ far_out) {
  __shared__ float dist[1024];
  __shared__ float rv[32];
  __shared__ int   ri[32];
  __shared__ int   s_cur;
  int t = threadIdx.x;
  if (t < n) dist[t] = 1e10f;
  if (t == 0) { s_cur = 0; far_out[0] = 0; }
  __syncthreads();
  for (int it = 1; it < npoint; ++it) {
    int cur = s_cur;
    float cx = xyz[3 * cur], cy = xyz[3 * cur + 1], cz = xyz[3 * cur + 2];
    float v = -1.0f; int vi = 0;
    if (t < n) {
      float dx = xyz[3 * t] - cx, dy = xyz[3 * t + 1] - cy, dz = xyz[3 * t + 2] - cz;
      float d = dx * dx + dy * dy + dz * dz;
      float nd = fminf(dist[t], d);
      dist[t] = nd; v = nd; vi = t;
    }
    for (int off = 16; off > 0; off >>= 1) {
      float ov = __shfl_down(v, off, 32);
      int   oi = __shfl_down(vi, off, 32);
      if (ov > v || (ov == v && oi < vi)) { v = ov; vi = oi; }
    }
    int wid = t >> 5;
    if ((t & 31) == 0) { rv[wid] = v; ri[wid] = vi; }
    __syncthreads();
    if (t == 0) {
      float bv = rv[0]; int bvi = ri[0];
      int nw = blockDim.x >> 5;
      for (int w = 1; w < nw; ++w)
        if (rv[w] > bv || (rv[w] == bv && ri[w] < bvi)) { bv = rv[w]; bvi = ri[w]; }
      s_cur = bvi; far_out[it] = bvi;
    }
    __syncthreads();
  }
}

// ---------------------------------------------------------------------------
// Small helpers
// ---------------------------------------------------------------------------
__global__ void gather_kernel(const float* __restrict__ src, const int* __restrict__ idx,
                              float* __restrict__ dst, int n, int c) {
  int i = blockIdx.x * blockDim.x + threadIdx.x;
  if (i >= n * c) return;
  int r = i / c, col = i % c;
  dst[i] = src[idx[r] * c + col];
}

__global__ void concat_kernel(const float* __restrict__ a, const float* __restrict__ b,
                              float* __restrict__ o, int n, int ca, int cb) {
  int ct = ca + cb;
  int i = blockIdx.x * blockDim.x + threadIdx.x;
  if (i >= n * ct) return;
  int r = i / ct, c = i % ct;
  o[i] = (c < ca) ? a[r * ca + c] : b[r * cb + (c - ca)];
}

// vfeat = relu([feat,xyz] @ W + b)
__global__ void vlinear_kernel(const float* __restrict__ feat, const float* __restrict__ xyz,
                               int ind, const float* __restrict__ W, const float* __restrict__ b,
                               float* __restrict__ out, int n, int outd) {
  int i = blockIdx.x * blockDim.x + threadIdx.x;
  if (i >= n * outd) return;
  int r = i / outd, o = i % outd;
  float acc = b[o];
  for (int j = 0; j < ind; ++j) acc += feat[r * ind + j] * W[j * outd + o];
  for (int j = 0; j < 3; ++j)   acc += xyz[r * 3 + j] * W[(ind + j) * outd + o];
  out[i] = fmaxf(acc, 0.0f);
}

__global__ void linear_kernel(const float* __restrict__ in, const float* __restrict__ W,
                              const float* __restrict__ b, float* __restrict__ out,
                              int n, int cin, int cout, int dorelu) {
  int i = blockIdx.x * blockDim.x + threadIdx.x;
  if (i >= n * cout) return;
  int r = i / cout, o = i % cout;
  float acc = b[o];
  for (int j = 0; j < cin; ++j) acc += in[r * cin + j] * W[j * cout + o];
  out[i] = dorelu ? fmaxf(acc, 0.0f) : acc;
}

__global__ void bn_stats_kernel(const float* __restrict__ x, int n, int c,
                                float* __restrict__ stats) {
  __shared__ float as[256], bs[256];
  int ch = blockIdx.x, t = threadIdx.x;
  float s = 0.0f, ss = 0.0f;
  for (int i = t; i < n; i += blockDim.x) {
    float v = x[i * c + ch];
    s += v; ss += v * v;
  }
  as[t] = s; bs[t] = ss;
  __syncthreads();
  for (int off = blockDim.x >> 1; off > 0; off >>= 1) {
    if (t < off) { as[t] += as[t + off]; bs[t] += bs[t + off]; }
    __syncthreads();
  }
  if (t == 0) {
    float mu = as[0] / (float)n;
    float var = bs[0] / (float)n - mu * mu;
    stats[ch] = mu;
    stats[c + ch] = rsqrtf(var + 1e-5f);
  }
}

__global__ void bn_apply_kernel(const float* __restrict__ x, const float* __restrict__ stats,
                                const float* __restrict__ g, const float* __restrict__ bb,
                                float* __restrict__ out, int n, int c) {
  int i = blockIdx.x * blockDim.x + threadIdx.x;
  if (i >= n * c) return;
  int ch = i % c;
  out[i] = (x[i] - stats[ch]) * stats[c + ch] * g[ch] + bb[ch];
}

// ---------------------------------------------------------------------------
// Fused transformer core (the r-GEMM + row-norm + gv einsum), f32 WMMA.
// 1 workgroup = 16 points, 8 waves; wave owns OUTD/8 c-values; accumulates
// its partial 16xOUTD out-tile in v8f registers across all (k,c).
// ---------------------------------------------------------------------------
template<int IND, int OUTD, int KNN>
__global__ __launch_bounds__(256) void tl_wmma_kernel(
    const float* __restrict__ feat, const float* __restrict__ xyz,
    const int* __restrict__ knn_idx,
    const float* __restrict__ Rw, const float* __restrict__ Rb,
    const float* __restrict__ vfeat, float* __restrict__ out, int N) {
  (void)N;
  constexpr int IN2 = 2 * (IND + 3);
  constexpr int K4  = (IN2 + 3) & ~3;
  constexpr int T   = OUTD / 16;   // col tiles covering o = 0..OUTD-1
  constexpr int CPW = OUTD / 8;    // c values per wave
  constexpr int LDB = OUTD * OUTD;

  __shared__ float Ash[16][K4];
  __shared__ int   nbr[16];
  __shared__ float gvk[16][OUTD];
  __shared__ float outAcc[16 * OUTD];

  const int n0     = blockIdx.x * 16;
  const int tid    = threadIdx.x;
  const int wave   = tid >> 5;
  const int lane   = tid & 31;
  const int lrow   = lane & 15;
  const int khalf  = (lane >> 4) << 1;  // 0 or 2 (A/B K-halves)
  const int rowoff = (lane >> 4) << 3;  // 0 or 8 (D rows)
  const float sq = sqrtf((float)OUTD);

  for (int i = tid; i < 16 * OUTD; i += 256) outAcc[i] = 0.0f;

  v8f acc[T];
#pragma unroll
  for (int t = 0; t < T; ++t) acc[t] = v8f_splat(0.0f);

  for (int k = 0; k < KNN; ++k) {
    __syncthreads();
    if (tid < 16) nbr[tid] = knn_idx[(n0 + tid) * KNN + k];
    __syncthreads();
    // build A = g rows: [pw[nbr]-pw[n], pw[n]]
    for (int e = tid; e < 16 * IN2; e += 256) {
      int m = e / IN2, d = e % IN2;
      int n = n0 + m, j = nbr[m];
      float val;
      if (d < IND + 3) {
        float pj = (d < IND) ? feat[j * IND + d] : xyz[j * 3 + (d - IND)];
        float pn = (d < IND) ? feat[n * IND + d] : xyz[n * 3 + (d - IND)];
        val = pj - pn;
      } else {
        int dd = d - (IND + 3);
        val = (dd < IND) ? feat[n * IND + dd] : xyz[n * 3 + (dd - IND)];
      }
      Ash[m][d] = val;
    }
    if constexpr (K4 > IN2) {
      for (int e = tid; e < 16 * (K4 - IN2); e += 256)
        Ash[e / (K4 - IN2)][IN2 + e % (K4 - IN2)] = 0.0f;
    }
    for (int e = tid; e < 16 * OUTD; e += 256)
      gvk[e / OUTD][e % OUTD] = vfeat[nbr[e / OUTD] * OUTD + e % OUTD];
    __syncthreads();

    for (int cc = 0; cc < CPW; ++cc) {
      const int c = wave * CPW + cc;
      const int col0 = c * OUTD;
      v8f dt[T];
#pragma unroll
      for (int t = 0; t < T; ++t) dt[t] = v8f_splat(Rb[col0 + t * 16 + lrow]);
      for (int ks = 0; ks < K4 / 4; ++ks) {
        const int k0 = ks * 4;
        int ra = k0 + khalf, rb = ra + 1;
        v2f a; a.x = Ash[lrow][ra]; a.y = Ash[lrow][rb];
        if (ra >= IN2) ra = IN2 - 1;   // padded rows: A is zero, keep B in-bounds
        if (rb >= IN2) rb = IN2 - 1;
#pragma unroll
        for (int t = 0; t < T; ++t) {
          const int col = col0 + t * 16 + lrow;
          v2f b; b.x = Rw[ra * LDB + col]; b.y = Rw[rb * LDB + col];
          dt[t] = wmma16x16x4(a, b, dt[t]);
        }
      }
      // row |.|-sums over the OUTD columns (cross-lane within 16-lane halves)
      float p[8];
#pragma unroll
      for (int j = 0; j < 8; ++j) {
        float s = 0.0f;
#pragma unroll
        for (int t = 0; t < T; ++t) s += fabsf(dt[t][j]);
        p[j] = s;
      }
#pragma unroll
      for (int j = 0; j < 8; ++j) {
        p[j] += __shfl_xor(p[j], 1, 32);
        p[j] += __shfl_xor(p[j], 2, 32);
        p[j] += __shfl_xor(p[j], 4, 32);
        p[j] += __shfl_xor(p[j], 8, 32);
      }
#pragma unroll
      for (int j = 0; j < 8; ++j) {
        const int m = j + rowoff;
        const float scale = gvk[m][c] * sq / (p[j] + (float)OUTD * 1e-7f);
#pragma unroll
        for (int t = 0; t < T; ++t) acc[t][j] += scale * dt[t][j];
      }
    }
  }
  __syncthreads();
#pragma unroll
  for (int t = 0; t < T; ++t)
#pragma unroll
    for (int j = 0; j < 8; ++j)
      atomicAdd(&outAcc[(j + rowoff) * OUTD + t * 16 + lrow], acc[t][j]);
  __syncthreads();
  for (int i = tid; i < 16 * OUTD; i += 256)
    out[(n0 + i / OUTD) * OUTD + i % OUTD] = outAcc[i];
}

// ---------------------------------------------------------------------------
// Indicator: h=relu(gx@W1+b1) in LDS, posw=h@W2+b2 via f32 WMMA, fused with
// the gf product and k-reduction (1 wg = 16 queries x 12 neighbors = 192 rows)
// ---------------------------------------------------------------------------
template<int OUTD>
__global__ __launch_bounds__(256) void indicator_wmma_kernel(
    const float* __restrict__ fpre, const float* __restrict__ xyz1,
    const float* __restrict__ xyz2, const int* __restrict__ idx,
    const float* __restrict__ W1, const float* __restrict__ b1,
    const float* __restrict__ W2, const float* __restrict__ b2,
    float* __restrict__ nf, int N2) {
  (void)N2;
  constexpr int KNN = 12, ROWS = 192, NT = OUTD / 16, MT = ROWS / 16;
  extern __shared__ float smem[];
  float* H     = smem;                 // [ROWS * OUTD]
  float* nfAcc = H + ROWS * OUTD;      // [16 * OUTD]
  int*   gidx  = (int*)(nfAcc + 16 * OUTD);  // [ROWS]

  const int n0     = blockIdx.x * 16;
  const int tid    = threadIdx.x;
  const int wave   = tid >> 5;
  const int lane   = tid & 31;
  const int lrow   = lane & 15;
  const int khalf  = (lane >> 4) << 1;
  const int rowoff = (lane >> 4) << 3;

  for (int i = tid; i < 16 * OUTD; i += 256) nfAcc[i] = 0.0f;
  for (int r = tid; r < ROWS; r += 256)
    gidx[r] = idx[(n0 + r / KNN) * KNN + r % KNN];
  __syncthreads();

  for (int e = tid; e < ROWS * OUTD; e += 256) {
    int r = e / OUTD, c = e % OUTD;
    int n2 = n0 + r / KNN, j = gidx[r];
    float gx = xyz1[3 * j]     - xyz2[3 * n2];
    float gy = xyz1[3 * j + 1] - xyz2[3 * n2 + 1];
    float gz = xyz1[3 * j + 2] - xyz2[3 * n2 + 2];
    float a = b1[c] + gx * W1[c] + gy * W1[OUTD + c] + gz * W1[2 * OUTD + c];
    H[e] = fmaxf(a, 0.0f);
  }
  __syncthreads();

  const float rs = 0.28867513459481287f;  // 1/sqrt(12)
  for (int tile = wave; tile < MT * NT; tile += 8) {
    const int mt = tile / NT, ct = tile % NT;
    const int col = ct * 16 + lrow;
    v8f dt = v8f_splat(b2[col]);
    for (int ks = 0; ks < OUTD / 4; ++ks) {
      const int k0 = ks * 4;
      const int arow = mt * 16 + lrow;
      v2f a; a.x = H[arow * OUTD + k0 + khalf]; a.y = H[arow * OUTD + k0 + khalf + 1];
      v2f b; b.x = W2[(k0 + khalf) * OUTD + col]; b.y = W2[(k0 + khalf + 1) * OUTD + col];
      dt = wmma16x16x4(a, b, dt);
    }
#pragma unroll
    for (int j = 0; j < 8; ++j) {
      const int r = mt * 16 + j + rowoff;
      const int m = r / KNN;
      const float gf = fpre[gidx[r] * OUTD + col];
      atomicAdd(&nfAcc[m * OUTD + col], dt[j] * gf * rs);
    }
  }
  __syncthreads();
  for (int i = tid; i < 16 * OUTD; i += 256)
    nf[(n0 + i / OUTD) * OUTD + i % OUTD] = nfAcc[i];
}

// ---------------------------------------------------------------------------
// Head: occ = cls2(relu(cls1(nf))), plus weight from nearest-neighbor dist
// ---------------------------------------------------------------------------
__global__ void cls_kernel(const float* __restrict__ nf,
                           const float* __restrict__ W1, const float* __restrict__ b1,
                           const float* __restrict__ W2, const float* __restrict__ b2,
                           const float* __restrict__ d0, float* __restrict__ out, int m) {
  int i = blockIdx.x * blockDim.x + threadIdx.x;
  if (i >= m) return;
  float h[16];
#pragma unroll
  for (int o = 0; o < 16; ++o) h[o] = b1[o];
  for (int j = 0; j < 256; ++j) {
    float x = nf[i * 256 + j];
#pragma unroll
    for (int o = 0; o < 16; ++o) h[o] += x * W1[j * 16 + o];
  }
  float o0 = b2[0], o1 = b2[1];
#pragma unroll
  for (int j = 0; j < 16; ++j) {
    float x = fmaxf(h[j], 0.0f);
    o0 += x * W2[j * 2 + 0];
    o1 += x * W2[j * 2 + 1];
  }
  out[i * 2 + 0] = o0;
  out[i * 2 + 1] = o1;
  out[2 * m + i] = (d0[i] > 0.03f) ? 10.0f : 1.0f;
}

// ---------------------------------------------------------------------------
// Orchestration
// ---------------------------------------------------------------------------
extern "C" void kernel_launch(void* const* d_in, const int* in_sizes, int n_in,
                              void* d_out, int out_size, void* d_ws, size_t ws_size,
                              hipStream_t stream) {
  (void)in_sizes; (void)n_in; (void)out_size; (void)ws_size;
  const float* xyz = (const float*)d_in[0];   // [1,1024,3]
  const float* det = (const float*)d_in[1];   // [1,4096,3]
  auto F = [&](int i) { return (const float*)d_in[i]; };
  // leaf order: xyz, detect, normal_gt, tl1..tl5 (r.w,r.b,v.w,v.b,suf.w,suf.b,bn_g,bn_b),
  //             it5/ind3 (prefix.w,prefix.b,pos1.w,pos1.b,pos2.w,pos2.b), cls1.w/b, cls2.w/b
  const int TL1 = 3, TL2 = 11, TL3 = 19, TL4 = 27, TL5 = 35, IT5 = 43, IND3 = 49, CLS = 55;

  char* ws = (char*)d_ws; size_t off = 0;
  auto alloc = [&](size_t bytes) -> void* {
    void* p = ws + off; off += (bytes + 255) & ~(size_t)255; return p;
  };
  int*   idx36   = (int*)alloc(1024 * 36 * 4);
  int*   idx36b  = (int*)alloc(512 * 36 * 4);
  int*   idx12a  = (int*)alloc(1024 * 12 * 4);
  int*   idx12b  = (int*)alloc(4096 * 12 * 4);
  float* d0b     = (float*)alloc(4096 * 4);
  float* vfeat   = (float*)alloc(1024 * 64 * 4);
  float* tout    = (float*)alloc(1024 * 64 * 4);
  float* tsuf    = (float*)alloc(1024 * 64 * 4);
  float* stats   = (float*)alloc(2 * 64 * 4);
  float* f1      = (float*)alloc(1024 * 16 * 4);
  float* f2      = (float*)alloc(1024 * 64 * 4);
  float* f3      = (float*)alloc(1024 * 64 * 4);
  float* f4      = (float*)alloc(512 * 64 * 4);
  float* f4up    = (float*)alloc(1024 * 64 * 4);
  float* f5      = (float*)alloc(1024 * 64 * 4);
  int*   far     = (int*)alloc(512 * 4);
  float* xyz512  = (float*)alloc(512 * 3 * 4);
  float* f3g     = (float*)alloc(512 * 64 * 4);
  float* fpre5   = (float*)alloc(512 * 64 * 4);
  float* featcat = (float*)alloc(1024 * 128 * 4);
  float* fpre3   = (float*)alloc(1024 * 256 * 4);
  float* nf      = (float*)alloc(4096 * 256 * 4);

  knn_kernel<36><<<CDIV(1024, 64), 64, 0, stream>>>(xyz, 1024, xyz, 1024, idx36, nullptr);

  // tl1: feat = xyz (IND=3, OUTD=16)
  vlinear_kernel<<<CDIV(1024 * 16, 256), 256, 0, stream>>>(xyz, xyz, 3, F(TL1 + 2), F(TL1 + 3), vfeat, 1024, 16);
  tl_wmma_kernel<3, 16, 36><<<64, 256, 0, stream>>>(xyz, xyz, idx36, F(TL1 + 0), F(TL1 + 1), vfeat, tout, 1024);
  linear_kernel<<<CDIV(1024 * 16, 256), 256, 0, stream>>>(tout, F(TL1 + 4), F(TL1 + 5), tsuf, 1024, 16, 16, 0);
  bn_stats_kernel<<<16, 256, 0, stream>>>(tsuf, 1024, 16, stats);
  bn_apply_kernel<<<CDIV(1024 * 16, 256), 256, 0, stream>>>(tsuf, stats, F(TL1 + 6), F(TL1 + 7), f1, 1024, 16);

  // tl2 (IND=16, OUTD=64)
  vlinear_kernel<<<CDIV(1024 * 64, 256), 256, 0, stream>>>(f1, xyz, 16, F(TL2 + 2), F(TL2 + 3), vfeat, 1024, 64);
  tl_wmma_kernel<16, 64, 36><<<64, 256, 0, stream>>>(f1, xyz, idx36, F(TL2 + 0), F(TL2 + 1), vfeat, tout, 1024);
  linear_kernel<<<CDIV(1024 * 64, 256), 256, 0, stream>>>(tout, F(TL2 + 4), F(TL2 + 5), tsuf, 1024, 64, 64, 0);
  bn_stats_kernel<<<64, 256, 0, stream>>>(tsuf, 1024, 64, stats);
  bn_apply_kernel<<<CDIV(1024 * 64, 256), 256, 0, stream>>>(tsuf, stats, F(TL2 + 6), F(TL2 + 7), f2, 1024, 64);

  // tl3 (IND=64, OUTD=64)
  vlinear_kernel<<<CDIV(1024 * 64, 256), 256, 0, stream>>>(f2, xyz, 64, F(TL3 + 2), F(TL3 + 3), vfeat, 1024, 64);
  tl_wmma_kernel<64, 64, 36><<<64, 256, 0, stream>>>(f2, xyz, idx36, F(TL3 + 0), F(TL3 + 1), vfeat, tout, 1024);
  linear_kernel<<<CDIV(1024 * 64, 256), 256, 0, stream>>>(tout, F(TL3 + 4), F(TL3 + 5), tsuf, 1024, 64, 64, 0);
  bn_stats_kernel<<<64, 256, 0, stream>>>(tsuf, 1024, 64, stats);
  bn_apply_kernel<<<CDIV(1024 * 64, 256), 256, 0, stream>>>(tsuf, stats, F(TL3 + 6), F(TL3 + 7), f3, 1024, 64);

  // FPS + gathers + knn on the 512 subset
  fps_kernel<<<1, 1024, 0, stream>>>(xyz, 1024, 512, far);
  gather_kernel<<<CDIV(512 * 3, 256), 256, 0, stream>>>(xyz, far, xyz512, 512, 3);
  gather_kernel<<<CDIV(512 * 64, 256), 256, 0, stream>>>(f3, far, f3g, 512, 64);
  knn_kernel<36><<<CDIV(512, 64), 64, 0, stream>>>(xyz512, 512, xyz512, 512, idx36b, nullptr);

  // tl4 on 512 points
  vlinear_kernel<<<CDIV(512 * 64, 256), 256, 0, stream>>>(f3g, xyz512, 64, F(TL4 + 2), F(TL4 + 3), vfeat, 512, 64);
  tl_wmma_kernel<64, 64, 36><<<32, 256, 0, stream>>>(f3g, xyz512, idx36b, F(TL4 + 0), F(TL4 + 1), vfeat, tout, 512);
  linear_kernel<<<CDIV(512 * 64, 256), 256, 0, stream>>>(tout, F(TL4 + 4), F(TL4 + 5), tsuf, 512, 64, 64, 0);
  bn_stats_kernel<<<64, 256, 0, stream>>>(tsuf, 512, 64, stats);
  bn_apply_kernel<<<CDIV(512 * 64, 256), 256, 0, stream>>>(tsuf, stats, F(TL4 + 6), F(TL4 + 7), f4, 512, 64);

  // it5 indicator: upsample 512 -> 1024
  linear_kernel<<<CDIV(512 * 64, 256), 256, 0, stream>>>(f4, F(IT5 + 0), F(IT5 + 1), fpre5, 512, 64, 64, 0);
  knn_kernel<12><<<CDIV(1024, 64), 64, 0, stream>>>(xyz512, 512, xyz, 1024, idx12a, nullptr);
  {
    size_t smem = (size_t)(192 * 64 + 16 * 64) * 4 + 192 * 4;
    indicator_wmma_kernel<64><<<64, 256, smem, stream>>>(fpre5, xyz512, xyz, idx12a,
        F(IT5 + 2), F(IT5 + 3), F(IT5 + 4), F(IT5 + 5), f4up, 1024);
  }

  // tl5 (IND=64, OUTD=64)
  vlinear_kernel<<<CDIV(1024 * 64, 256), 256, 0, stream>>>(f4up, xyz, 64, F(TL5 + 2), F(TL5 + 3), vfeat, 1024, 64);
  tl_wmma_kernel<64, 64, 36><<<64, 256, 0, stream>>>(f4up, xyz, idx36, F(TL5 + 0), F(TL5 + 1), vfeat, tout, 1024);
  linear_kernel<<<CDIV(1024 * 64, 256), 256, 0, stream>>>(tout, F(TL5 + 4), F(TL5 + 5), tsuf, 1024, 64, 64, 0);
  bn_stats_kernel<<<64, 256, 0, stream>>>(tsuf, 1024, 64, stats);
  bn_apply_kernel<<<CDIV(1024 * 64, 256), 256, 0, stream>>>(tsuf, stats, F(TL5 + 6), F(TL5 + 7), f5, 1024, 64);

  // feat = [f3, f5]; ind3 indicator onto detect points
  concat_kernel<<<CDIV(1024 * 128, 256), 256, 0, stream>>>(f3, f5, featcat, 1024, 64, 64);
  linear_kernel<<<CDIV(1024 * 256, 256), 256, 0, stream>>>(featcat, F(IND3 + 0), F(IND3 + 1), fpre3, 1024, 128, 256, 0);
  knn_kernel<12><<<CDIV(4096, 64), 64, 0, stream>>>(xyz, 1024, det, 4096, idx12b, d0b);
  {
    size_t smem = (size_t)(192 * 256 + 16 * 256) * 4 + 192 * 4;
    indicator_wmma_kernel<256><<<256, 256, smem, stream>>>(fpre3, xyz, det, idx12b,
        F(IND3 + 2), F(IND3 + 3), F(IND3 + 4), F(IND3 + 5), nf, 4096);
  }

  // head: occ [4096,2] then weight [4096]
  cls_kernel<<<CDIV(4096, 128), 128, 0, stream>>>(nf, F(CLS + 0), F(CLS + 1),
      F(CLS + 2), F(CLS + 3), d0b, (float*)d_out, 4096);
}